// GCN_21208548508026
// MI455X (gfx1250) — compile-verified
//
#include <hip/hip_runtime.h>
#include <hip/hip_bf16.h>

typedef float v2f __attribute__((ext_vector_type(2)));
typedef float v8f __attribute__((ext_vector_type(8)));

#define NN    50000
#define EE    800000
#define GG    500
#define DIN   78
#define DINP  80
#define DH    312
#define DHP   320
#define BNEPS 1e-5f

// ---------------- utility fills ----------------
__global__ void fill_f32(float* __restrict__ p, float v, long n) {
  long i = (long)blockIdx.x * blockDim.x + threadIdx.x;
  if (i < n) p[i] = v;
}
__global__ void fill_u32(unsigned* __restrict__ p, unsigned v, long n) {
  long i = (long)blockIdx.x * blockDim.x + threadIdx.x;
  if (i < n) p[i] = v;
}

// ---------------- degree / norm ----------------
__global__ void deg_count(const int* __restrict__ dst, float* __restrict__ deg, int E) {
  int e = blockIdx.x * blockDim.x + threadIdx.x;
  if (e < E) atomicAdd(&deg[dst[e]], 1.0f);
}
__global__ void rsqrt_inplace(float* __restrict__ p, int n) {
  int i = blockIdx.x * blockDim.x + threadIdx.x;
  if (i < n) p[i] = rsqrtf(p[i]);
}

// ---------------- padding ----------------
__global__ void pad_x(const float* __restrict__ x, float* __restrict__ xp, int M) {
  int n = blockIdx.x, c = threadIdx.x;
  if (n >= M || c >= DINP) return;
  xp[(size_t)n * DINP + c] = (c < DIN) ? x[(size_t)n * DIN + c] : 0.0f;
}
__global__ void pad_W(const float* __restrict__ W, float* __restrict__ Wp, int K, int Nn) {
  int r = blockIdx.x, c = threadIdx.x;  // grid = Kpad rows, block = DHP
  Wp[(size_t)r * DHP + c] = (r < K && c < Nn) ? W[(size_t)r * Nn + c] : 0.0f;
}

// ---------------- f32 WMMA GEMM: C[M,Npad] = A[M,Kpad] * B[Kpad,Npad] ----------------
// One wave computes one 16x16 C tile via V_WMMA_F32_16X16X4_F32.
// A 16x4 frag: lane m=lane&15; vgpr0 = A[m][k+2*half], vgpr1 = A[m][k+2*half+1]
// B 4x16 frag: lane n=lane&15; vgpr0 = B[k+2*half][n], vgpr1 = B[k+2*half+1][n]
// C 16x16:     vgpr r -> row (half ? r+8 : r), col = lane&15
__global__ __launch_bounds__(256) void gemm_f32_wmma(
    const float* __restrict__ A, int lda,
    const float* __restrict__ B,
    float* __restrict__ C,
    int M, int Npad, int Kpad) {
  int wave = (blockIdx.x * blockDim.x + threadIdx.x) >> 5;
  int lane = threadIdx.x & 31;
  int ntn  = Npad >> 4;
  int tm = wave / ntn;
  int tn = wave - tm * ntn;
  if (tm * 16 >= M) return;  // wave-uniform: EXEC stays all-ones for WMMA

  int half = lane >> 4;
  int m    = lane & 15;
  int row0 = tm * 16;
  int arow = row0 + m;
  const float* Arow = A + (size_t)(arow < M ? arow : (M - 1)) * lda;
  const float* Bcol = B + (tn * 16 + m);

  v8f acc = {0.f, 0.f, 0.f, 0.f, 0.f, 0.f, 0.f, 0.f};
#pragma unroll 4
  for (int k = 0; k < Kpad; k += 4) {
    int ka = k + half * 2;
    v2f a, b;
    a.x = Arow[ka];
    a.y = Arow[ka + 1];
    b.x = Bcol[(size_t)ka * DHP];
    b.y = Bcol[(size_t)(ka + 1) * DHP];
    acc = __builtin_amdgcn_wmma_f32_16x16x4_f32(false, a, false, b,
                                                (short)0, acc, false, false);
  }
  int col = tn * 16 + m;
#pragma unroll
  for (int r = 0; r < 8; ++r) {
    int row = row0 + r + half * 8;
    if (row < M) C[(size_t)row * DHP + col] = acc[r];
  }
}

// ---------------- GCN aggregation ----------------
// self-loop term: agg[n] = h[n] * dinv[n]^2   (pad cols stay 0 since h pad cols are 0)
__global__ void agg_init(const float* __restrict__ h, const float* __restrict__ dinv,
                         float* __restrict__ agg, int M) {
  int n = blockIdx.x, c = threadIdx.x;  // block = DHP
  if (n >= M) return;
  float di = dinv[n];
  agg[(size_t)n * DHP + c] = h[(size_t)n * DHP + c] * di * di;
}
// edge scatter: agg[d] += h[s] * dinv[s]*dinv[d]  (L2-resident: h buf = 64MB < 192MB L2)
__global__ void scatter_edges(const float* __restrict__ h, const int* __restrict__ src,
                              const int* __restrict__ dst, const float* __restrict__ dinv,
                              float* __restrict__ agg, int E) {
  int e = blockIdx.x, c = threadIdx.x;  // block = DHP
  if (e >= E || c >= DH) return;
  int s = src[e], d = dst[e];
  float norm = dinv[s] * dinv[d];
  atomicAdd(&agg[(size_t)d * DHP + c], h[(size_t)s * DHP + c] * norm);
}

// ---------------- bias + ReLU + BN moment accumulation (in place) ----------------
__global__ void bias_relu_stats(float* __restrict__ y, const float* __restrict__ bias,
                                float* __restrict__ sums, int M) {
  int c = threadIdx.x;  // block = DHP
  if (c >= DH) return;
  int r0 = blockIdx.x * 128;
  int r1 = min(r0 + 128, M);
  float b = bias[c], s = 0.f, s2 = 0.f;
  for (int r = r0; r < r1; ++r) {
    float v = y[(size_t)r * DHP + c] + b;
    v = fmaxf(v, 0.0f);
    y[(size_t)r * DHP + c] = v;
    s += v;
    s2 += v * v;
  }
  atomicAdd(&sums[c], s);
  atomicAdd(&sums[DHP + c], s2);
}
__global__ void bn_apply(float* __restrict__ y, const float* __restrict__ sums,
                         const float* __restrict__ gamma, const float* __restrict__ beta,
                         int M) {
  int c = threadIdx.x;  // block = DHP
  if (c >= DH) return;
  int r0 = blockIdx.x * 128;
  int r1 = min(r0 + 128, M);
  float invn  = 1.0f / (float)M;
  float mean  = sums[c] * invn;
  float var   = sums[DHP + c] * invn - mean * mean;
  float scale = rsqrtf(var + BNEPS) * gamma[c];
  float shift = beta[c] - mean * scale;
  for (int r = r0; r < r1; ++r)
    y[(size_t)r * DHP + c] = y[(size_t)r * DHP + c] * scale + shift;
}

// ---------------- global max pool (monotonic uint encoding) ----------------
__device__ __forceinline__ unsigned enc_f32(float f) {
  unsigned u = __float_as_uint(f);
  return (u & 0x80000000u) ? ~u : (u | 0x80000000u);
}
__device__ __forceinline__ float dec_f32(unsigned u) {
  return __uint_as_float((u & 0x80000000u) ? (u & 0x7FFFFFFFu) : ~u);
}
__global__ void pool_max(const float* __restrict__ y, const int* __restrict__ batch,
                         unsigned* __restrict__ pool, int M) {
  int n = blockIdx.x, c = threadIdx.x;  // block = DHP
  if (n >= M || c >= DH) return;
  int g = batch[n];
  atomicMax(&pool[(size_t)g * DH + c], enc_f32(y[(size_t)n * DHP + c]));
}
__global__ void pool_decode(const unsigned* __restrict__ pool, float* __restrict__ out) {
  int g = blockIdx.x, c = threadIdx.x;
  if (c >= DH) return;
  out[(size_t)g * DH + c] = dec_f32(pool[(size_t)g * DH + c]);
}

// ---------------- orchestration ----------------
extern "C" void kernel_launch(void* const* d_in, const int* in_sizes, int n_in,
                              void* d_out, int out_size, void* d_ws, size_t ws_size,
                              hipStream_t stream) {
  const float* x      = (const float*)d_in[0];
  const int*   eidx   = (const int*)d_in[1];
  const int*   batch  = (const int*)d_in[2];
  const float* W1     = (const float*)d_in[3];
  const float* b1     = (const float*)d_in[4];
  const float* gamma1 = (const float*)d_in[5];
  const float* beta1  = (const float*)d_in[6];
  const float* W2     = (const float*)d_in[7];
  const float* b2     = (const float*)d_in[8];
  const float* gamma2 = (const float*)d_in[9];
  const float* beta2  = (const float*)d_in[10];
  float* out = (float*)d_out;

  const int* src = eidx;        // edge_index[0,:]
  const int* dst = eidx + EE;   // edge_index[1,:]

  // workspace carve-up (offsets in floats, 32-float aligned)
  float* ws = (float*)d_ws;
  float*    deg  = ws;                         // N                  -> dinv in place
  float*    xp   = ws + 50048;                 // N*DINP = 4,000,000
  float*    Wp1  = ws + 4050048;               // DINP*DHP = 25,600
  float*    Wp2  = ws + 4075648;               // DHP*DHP  = 102,400
  float*    sums = ws + 4178048;               // 2*DHP = 640
  unsigned* pool = (unsigned*)(ws + 4178688);  // G*DH = 156,000
  float*    bufA = ws + 4334688;               // N*DHP = 16,000,000 (h)
  float*    bufB = ws + 20334688;              // N*DHP = 16,000,000 (agg/z)

  const int statBlocks = (NN + 127) / 128;
  const int gemmTiles  = (NN / 16) * (DHP / 16);          // 3125 * 20
  const int gemmGrid   = (gemmTiles + 7) / 8;             // 8 waves / 256-thread block

  // degree + D^{-1/2}
  fill_f32<<<(NN + 255) / 256, 256, 0, stream>>>(deg, 1.0f, NN);
  deg_count<<<(EE + 255) / 256, 256, 0, stream>>>(dst, deg, EE);
  rsqrt_inplace<<<(NN + 255) / 256, 256, 0, stream>>>(deg, NN);

  // pad inputs/weights
  pad_x<<<NN, 128, 0, stream>>>(x, xp, NN);
  pad_W<<<DINP, DHP, 0, stream>>>(W1, Wp1, DIN, DH);
  pad_W<<<DHP, DHP, 0, stream>>>(W2, Wp2, DH, DH);

  // ---- layer 1 ----
  gemm_f32_wmma<<<gemmGrid, 256, 0, stream>>>(xp, DINP, Wp1, bufA, NN, DHP, DINP);
  agg_init<<<NN, DHP, 0, stream>>>(bufA, deg, bufB, NN);
  scatter_edges<<<EE, DHP, 0, stream>>>(bufA, src, dst, deg, bufB, EE);
  fill_f32<<<(2 * DHP + 255) / 256, 256, 0, stream>>>(sums, 0.0f, 2 * DHP);
  bias_relu_stats<<<statBlocks, DHP, 0, stream>>>(bufB, b1, sums, NN);
  bn_apply<<<statBlocks, DHP, 0, stream>>>(bufB, sums, gamma1, beta1, NN);

  // ---- layer 2 ----  (bufB = z1 is GEMM input; safely reused as agg2 afterwards)
  gemm_f32_wmma<<<gemmGrid, 256, 0, stream>>>(bufB, DHP, Wp2, bufA, NN, DHP, DHP);
  agg_init<<<NN, DHP, 0, stream>>>(bufA, deg, bufB, NN);
  scatter_edges<<<EE, DHP, 0, stream>>>(bufA, src, dst, deg, bufB, EE);
  fill_f32<<<(2 * DHP + 255) / 256, 256, 0, stream>>>(sums, 0.0f, 2 * DHP);
  bias_relu_stats<<<statBlocks, DHP, 0, stream>>>(bufB, b2, sums, NN);
  bn_apply<<<statBlocks, DHP, 0, stream>>>(bufB, sums, gamma2, beta2, NN);

  // ---- global max pool ----
  fill_u32<<<((long)GG * DH + 255) / 256, 256, 0, stream>>>(pool, 0u, (long)GG * DH);
  pool_max<<<NN, DHP, 0, stream>>>(bufB, batch, pool, NN);
  pool_decode<<<GG, DHP, 0, stream>>>(pool, out);
}